// EMA1D_14113262535423
// MI455X (gfx1250) — compile-verified
//
#include <hip/hip_runtime.h>
#include <cstdint>

// EMA1D straight-through estimator, forward pass.
//
// reference: state = EMA(stop_grad(x)); return state - stop_grad(state) + x
// Forward value == x exactly (state - state == +0.0, +0.0 + x == x).
// => optimal kernel is an identity copy: 256 MB in + 256 MB out = 512 MB,
//    ~22 us at 23.3 TB/s. Zero FLOPs; the EMA scan is gradient-only dead code.
//
// Implementation: CDNA5 async copy path. Each wave DMA-streams 2 KB tiles
// through LDS with GLOBAL_LOAD_ASYNC_TO_LDS_B128 / GLOBAL_STORE_ASYNC_FROM_LDS_B128
// (ASYNCcnt-tracked, data never occupies VGPRs), double-buffered so the
// unordered async LDS accesses never overlap (load[t+1] writes buffer B while
// store[t] still reads buffer A; s_wait_asynccnt 0 fences each tile's loads
// and the previous tile's stores). ~4 KB in flight per wave x 8192 waves
// ≈ 32 MB outstanding — enough to cover HBM latency at 23.3 TB/s.

__global__ __launch_bounds__(256) void ema_st_identity_async_copy(
    const float* __restrict__ in, float* __restrict__ out, long long n)
{
    // 8 waves/block * (2 buffers * 2 KB) = 32 KB LDS
    __shared__ char smem[8 * 4096];

    const int lane  = threadIdx.x & 31;          // wave32
    const int waveb = threadIdx.x >> 5;

    const long long waves_per_grid = (long long)gridDim.x * (blockDim.x >> 5);
    const long long wave_gid = (long long)blockIdx.x * (blockDim.x >> 5) + waveb;

    // Generic LDS pointer low 32 bits == LDS byte address (aperture truncation).
    const uint32_t lds0 = ((uint32_t)(uintptr_t)&smem[0])
                        + (uint32_t)(waveb * 4096 + lane * 16);

    const long long n4     = n >> 2;     // float4 count
    const long long TILE4  = 128;        // float4 per wave-tile: 4 ops x 32 lanes (2 KB)
    const long long ntiles = n4 / TILE4;

    const float4* __restrict__ in4  = (const float4*)in;
    float4* __restrict__       out4 = (float4*)out;

    uint32_t parity = 0;
    for (long long t = wave_gid; t < ntiles; t += waves_per_grid) {
        const float4* gin  = in4  + t * TILE4 + lane;
        float4*       gout = out4 + t * TILE4 + lane;
        const uint32_t ldsa = lds0 + (parity << 11);   // toggle 2 KB buffer
        parity ^= 1u;

        // NOTE: instruction offset is added to BOTH the global and the LDS
        // address for async ops, so one pair of address regs drives 4 ops.
        asm volatile(
            "global_load_async_to_lds_b128 %0, %1, off\n\t"
            "global_load_async_to_lds_b128 %0, %1, off offset:512\n\t"
            "global_load_async_to_lds_b128 %0, %1, off offset:1024\n\t"
            "global_load_async_to_lds_b128 %0, %1, off offset:1536"
            :: "v"(ldsa), "v"(gin) : "memory");

        // Waits this tile's 4 loads AND the previous tile's 4 stores.
        asm volatile("s_wait_asynccnt 0x0" ::: "memory");

        asm volatile(
            "global_store_async_from_lds_b128 %0, %1, off\n\t"
            "global_store_async_from_lds_b128 %0, %1, off offset:512\n\t"
            "global_store_async_from_lds_b128 %0, %1, off offset:1024\n\t"
            "global_store_async_from_lds_b128 %0, %1, off offset:1536"
            :: "v"(gout), "v"(ldsa) : "memory");
    }
    // S_ENDPGM performs an implicit wait-idle (all counters -> 0), so the
    // trailing stores are guaranteed complete at kernel end.

    // Tail for n not a multiple of 512 floats (empty for 64*128*8192).
    const long long covered = ntiles * (TILE4 * 4);
    const long long gtid    = (long long)blockIdx.x * blockDim.x + threadIdx.x;
    const long long gstride = (long long)gridDim.x * blockDim.x;
    for (long long i = covered + gtid; i < n; i += gstride)
        out[i] = in[i];
}

extern "C" void kernel_launch(void* const* d_in, const int* in_sizes, int n_in,
                              void* d_out, int out_size, void* d_ws, size_t ws_size,
                              hipStream_t stream) {
    (void)n_in; (void)d_ws; (void)ws_size; (void)out_size;
    const float* x   = (const float*)d_in[0];
    float*       out = (float*)d_out;
    const long long n = (long long)in_sizes[0];   // 64*128*8192 = 67,108,864

    // 1024 blocks x 256 threads = 8192 waves; 16 wave-tiles each (grid-stride).
    ema_st_identity_async_copy<<<1024, 256, 0, stream>>>(x, out, n);
}